// OTBlock_87479893885023
// MI455X (gfx1250) — compile-verified
//
#include <hip/hip_runtime.h>

typedef float v2f __attribute__((ext_vector_type(2)));
typedef float v8f __attribute__((ext_vector_type(8)));

#define NUM_P   16384
#define DIM     100
#define BATN    8192
#define PSPLIT  4
#define P_PER_SPLIT   (NUM_P / PSPLIT)          // 4096
#define PT_PER_SPLIT  (P_PER_SPLIT / 16)        // 256 tiles of 16 rows
#define KSTEPS  (DIM / 4)                       // 25 WMMA K-steps
#define WAVES_PER_BLOCK 8
#define COLS_PER_BLOCK  (WAVES_PER_BLOCK * 16)  // 128 columns of N
#define NBLOCKS_X (BATN / COLS_PER_BLOCK)       // 64

// ---------------------------------------------------------------------------
// Main kernel: fp32 WMMA GEMM tile sweep + running (max, argmax) per column.
// grid = (64, PSPLIT), block = 256 (8 waves). Wave w of block bx owns columns
// n = (bx*8 + w)*16 .. +15 ; blockIdx.y selects a 4096-row slice of P.
// ---------------------------------------------------------------------------
__global__ __launch_bounds__(256)
void omt_wmma_argmax(const float* __restrict__ hP,
                     const float* __restrict__ h,
                     const float* __restrict__ volP,
                     float* __restrict__ part_val,
                     int*   __restrict__ part_idx)
{
    __shared__ float ldsA[16 * DIM];   // one 16-row A tile (h_P), 6.25 KB
    __shared__ float ldsH[16];         // matching slice of h

    const int tid  = threadIdx.x;
    const int wave = tid >> 5;
    const int lane = tid & 31;
    const int col  = lane & 15;        // N (and A-row M) sub-index
    const int half = lane >> 4;        // K-pair selector in A/B fragments

    const int n = (blockIdx.x * WAVES_PER_BLOCK + wave) * 16 + col;

    // ---- Preload this wave's entire B tile into registers (stays resident).
    // B fragment (4x16 f32): lane holds column N = lane%16, K = kbase+2*half..+1
    v2f breg[KSTEPS];
    const float* bp = volP + (size_t)n * DIM + 2 * half;
#pragma unroll
    for (int kk = 0; kk < KSTEPS; ++kk)
        breg[kk] = *(const v2f*)(bp + kk * 4);

    float bestv = -__builtin_inff();
    int   bestp = 0;

    const int p0base = blockIdx.y * P_PER_SPLIT;

    for (int t = 0; t < PT_PER_SPLIT; ++t) {
        const int p0 = p0base + t * 16;

        __syncthreads();   // protect LDS from previous iteration's readers
        // Cooperative fill: rows p0..p0+15 of h_P are one contiguous 1600-float
        // block; all 8 waves share it.
        for (int i = tid; i < 16 * DIM; i += 256)
            ldsA[i] = hP[(size_t)p0 * DIM + i];
        if (tid < 16) ldsH[tid] = h[p0 + tid];
        __syncthreads();

        // A fragment (16x4 f32): lane holds row M = lane%16, K = kbase+2*half..+1
        const float* arow = &ldsA[col * DIM + 2 * half];

        v8f c = {0.f, 0.f, 0.f, 0.f, 0.f, 0.f, 0.f, 0.f};
#pragma unroll
        for (int kk = 0; kk < KSTEPS; ++kk) {
            v2f a = *(const v2f*)(arow + kk * 4);
            c = __builtin_amdgcn_wmma_f32_16x16x4_f32(
                    /*neg_a=*/false, a, /*neg_b=*/false, breg[kk],
                    /*c_mod=*/(short)0, c, /*reuse_a=*/false, /*reuse_b=*/false);
        }

        // C layout: VGPR v holds row M = v + 8*half for this lane's column.
        // Ascending v + strict '>' keeps the first (lowest-p) maximum.
#pragma unroll
        for (int v = 0; v < 8; ++v) {
            const int   prow = v + 8 * half;
            const float val  = c[v] + ldsH[prow];
            if (val > bestv) { bestv = val; bestp = p0 + prow; }
        }
    }

    // Merge the two lane-halves covering the same column (rows 0-7 vs 8-15 of
    // each tile). Tie -> lower p index (first occurrence).
    const float ov = __shfl_xor(bestv, 16, 32);
    const int   op = __shfl_xor(bestp, 16, 32);
    if (ov > bestv || (ov == bestv && op < bestp)) { bestv = ov; bestp = op; }

    if (half == 0) {
        part_val[(size_t)blockIdx.y * BATN + n] = bestv;
        part_idx[(size_t)blockIdx.y * BATN + n] = bestp;
    }
}

// ---------------------------------------------------------------------------
__global__ void omt_zero_counts(int* __restrict__ counts)
{
    const int p = blockIdx.x * blockDim.x + threadIdx.x;
    if (p < NUM_P) counts[p] = 0;
}

// Merge PSPLIT partial (max, argmax) per column, write final outputs, bincount.
// Splits are in ascending-p order, so strict '>' with tie->lower-index exactly
// reproduces jnp.argmax first-occurrence semantics.
__global__ void omt_reduce(const float* __restrict__ part_val,
                           const int*   __restrict__ part_idx,
                           float* __restrict__ out_val,
                           int*   __restrict__ out_idx,
                           int*   __restrict__ counts)
{
    const int nn = blockIdx.x * blockDim.x + threadIdx.x;
    if (nn >= BATN) return;
    float best = part_val[nn];
    int   bi   = part_idx[nn];
#pragma unroll
    for (int s = 1; s < PSPLIT; ++s) {
        const float v  = part_val[(size_t)s * BATN + nn];
        const int   ix = part_idx[(size_t)s * BATN + nn];
        if (v > best || (v == best && ix < bi)) { best = v; bi = ix; }
    }
    out_val[nn] = best;
    out_idx[nn] = bi;
    atomicAdd(&counts[bi], 1);
}

__global__ void omt_normalize(const int* __restrict__ counts,
                              float* __restrict__ g)
{
    const int p = blockIdx.x * blockDim.x + threadIdx.x;
    if (p < NUM_P) g[p] = (float)counts[p] * (1.0f / (float)BATN);
}

// ---------------------------------------------------------------------------
extern "C" void kernel_launch(void* const* d_in, const int* in_sizes, int n_in,
                              void* d_out, int out_size, void* d_ws, size_t ws_size,
                              hipStream_t stream)
{
    const float* hP   = (const float*)d_in[0];   // [16384, 100]
    const float* h    = (const float*)d_in[1];   // [16384]
    const float* volP = (const float*)d_in[2];   // [8192, 100]

    // d_out: [ tot_ind_val f32 x8192 | tot_ind i32 x8192 | g f32 x16384 ]
    float* out_val = (float*)d_out;
    int*   out_idx = (int*)d_out + BATN;
    float* out_g   = (float*)d_out + 2 * BATN;

    // Workspace layout (deterministic, re-initialized every call):
    float* part_val = (float*)d_ws;                                   // PSPLIT*8192 f32
    int*   part_idx = (int*)((char*)d_ws + sizeof(float) * PSPLIT * BATN);
    int*   counts   = (int*)((char*)d_ws + 2 * sizeof(float) * PSPLIT * BATN);

    omt_zero_counts<<<(NUM_P + 255) / 256, 256, 0, stream>>>(counts);

    omt_wmma_argmax<<<dim3(NBLOCKS_X, PSPLIT), 256, 0, stream>>>(
        hP, h, volP, part_val, part_idx);

    omt_reduce<<<(BATN + 255) / 256, 256, 0, stream>>>(
        part_val, part_idx, out_val, out_idx, counts);

    omt_normalize<<<(NUM_P + 255) / 256, 256, 0, stream>>>(counts, out_g);
}